// GATWithAtt_49100066128551
// MI455X (gfx1250) — compile-verified
//
#include <hip/hip_runtime.h>
#include <hip/hip_bf16.h>
#include <math.h>
#include <stdint.h>

// ---------------------------------------------------------------------------
// GAT 2-layer inference for MI455X (gfx1250, wave32).
//  - GEMM1: v_wmma_f32_16x16x4_f32, 80-row blocks, B fragments reused across
//    5 row tiles, x tile staged via global_load_async_to_lds_b128 (ASYNCcnt).
//  - Everything else: HBM-bound segment-softmax / scatter-aggregate pipeline.
// ---------------------------------------------------------------------------

typedef __attribute__((ext_vector_type(2)))  float  v2f;
typedef __attribute__((ext_vector_type(8)))  float  v8f;

#define IN_DIM   128
#define HID      64
#define HEADS    4
#define F1       (HEADS * HID)   /* 256 */
#define OUT_DIM  2
#define NEG_SLOPE 0.2f
#define BN_EPS    1e-5f

#define ROWS_PB   80             /* rows per block: 5 row-tiles of 16 */
#define RT_TILES  5
#define LDS_STRIDE (IN_DIM + 4)  /* 132 dwords: 132 % 64 == 4 -> no bank conflicts */

// -------------------------------------------------------------- fill helper
__global__ void fill_kernel(float* __restrict__ p, long long n, float v) {
  long long t = (long long)blockIdx.x * blockDim.x + threadIdx.x;
  if (t < n) p[t] = v;
}

// -------------------------------------------------------------- GEMM1 (WMMA)
// h1[N,256] = x[N,128] @ W1[128,256].
// Block = 256 threads (8 waves) handles 80 rows x 256 cols.
// Each wave owns two 16-wide column tiles and 5 row tiles; B fragments are
// loaded once per k-step and reused across all 5 row tiles.
__global__ __launch_bounds__(256)
void gemm1_wmma_kernel(const float* __restrict__ x, const float* __restrict__ W1,
                       float* __restrict__ h1, int n) {
  __shared__ float lds_x[ROWS_PB][LDS_STRIDE];
  const int row_base = blockIdx.x * ROWS_PB;

  // ---- stage x tile: async global->LDS, 16B per lane per iteration --------
  for (int it = 0; it < (ROWS_PB * IN_DIM) / (256 * 4); ++it) {
    const int flat = (it * 256 + (int)threadIdx.x) * 4;   // float index
    const int r = flat >> 7;          // /128
    const int c = flat & 127;
    const int gr = row_base + r;
    if (gr < n) {
      uint64_t gaddr = (uint64_t)(const void*)(x + (size_t)gr * IN_DIM + c);
      uint32_t laddr = (uint32_t)(uintptr_t)&lds_x[r][c];
      asm volatile("global_load_async_to_lds_b128 %0, %1, off"
                   :: "v"(laddr), "v"(gaddr) : "memory");
    } else {
      float4* lp = (float4*)&lds_x[r][c];
      *lp = make_float4(0.f, 0.f, 0.f, 0.f);
    }
  }
  asm volatile("s_wait_asynccnt 0x0" ::: "memory");
  __syncthreads();

  const int wave = threadIdx.x >> 5;
  const int lane = threadIdx.x & 31;
  const int half = lane >> 4;       // 0: lanes 0-15 (K0,K1), 1: lanes 16-31 (K2,K3)
  const int l16  = lane & 15;
  const int c0 = wave * 32;         // first column tile
  const int c1 = wave * 32 + 16;    // second column tile

  v8f acc0[RT_TILES];
  v8f acc1[RT_TILES];
#pragma unroll
  for (int rt = 0; rt < RT_TILES; ++rt) { acc0[rt] = (v8f){}; acc1[rt] = (v8f){}; }

  for (int k0 = 0; k0 < IN_DIM; k0 += 4) {
    const int ka = k0 + half * 2;
    // B fragment (4x16, mirrored A layout): load once, reuse for 5 row tiles.
    const float* w0 = W1 + (size_t)ka * F1;
    v2f b0, b1;
    b0.x = w0[c0 + l16];        b0.y = w0[F1 + c0 + l16];
    b1.x = w0[c1 + l16];        b1.y = w0[F1 + c1 + l16];
#pragma unroll
    for (int rt = 0; rt < RT_TILES; ++rt) {
      v2f a;
      a.x = lds_x[rt * 16 + l16][ka];
      a.y = lds_x[rt * 16 + l16][ka + 1];
      acc0[rt] = __builtin_amdgcn_wmma_f32_16x16x4_f32(false, a, false, b0,
                                                       (short)0, acc0[rt], false, false);
      acc1[rt] = __builtin_amdgcn_wmma_f32_16x16x4_f32(false, a, false, b1,
                                                       (short)0, acc1[rt], false, false);
    }
  }

  // ---- store: C/D layout VGPR i -> M=i (lanes 0-15) / M=8+i (lanes 16-31) --
#pragma unroll
  for (int rt = 0; rt < RT_TILES; ++rt) {
    const int r0 = row_base + rt * 16 + half * 8;  // first of 8 rows this lane owns
    float* p = h1 + (size_t)r0 * F1 + c0 + l16;
    if (r0 + 7 < n) {                               // whole tile in bounds (uniform)
#pragma unroll
      for (int i = 0; i < 8; ++i) {
        p[(size_t)i * F1]      = acc0[rt][i];
        p[(size_t)i * F1 + 16] = acc1[rt][i];
      }
    } else {
#pragma unroll
      for (int i = 0; i < 8; ++i) {
        if (r0 + i < n) {
          p[(size_t)i * F1]      = acc0[rt][i];
          p[(size_t)i * F1 + 16] = acc1[rt][i];
        }
      }
    }
  }
}

// ------------------------------------------------- per-node attention logits
__global__ void att_logits1_kernel(const float* __restrict__ h1,
                                   const float* __restrict__ a_src,
                                   const float* __restrict__ a_dst,
                                   float* __restrict__ es, float* __restrict__ ed,
                                   int n) {
  int t = blockIdx.x * blockDim.x + threadIdx.x;
  if (t >= n * HEADS) return;
  const int node = t / HEADS, hd = t % HEADS;
  const float* hp = h1 + (size_t)node * F1 + hd * HID;
  const float* as = a_src + hd * HID;
  const float* ad = a_dst + hd * HID;
  float s = 0.f, d = 0.f;
  for (int c = 0; c < HID; ++c) { float v = hp[c]; s += v * as[c]; d += v * ad[c]; }
  es[t] = s;
  ed[t] = d;
}

// ------------------------------------- edge pass 1: leaky-relu + segment max
__global__ void edge_logits_max_kernel(const float* __restrict__ es,
                                       const float* __restrict__ ed,
                                       const int* __restrict__ ei,
                                       int E, int n, int H,
                                       float* __restrict__ e_buf,
                                       float* __restrict__ mbuf) {
  int t = blockIdx.x * blockDim.x + threadIdx.x;
  if (t >= (E + n) * H) return;
  const int e = t / H, hd = t % H;
  int s, d;
  if (e < E) { s = ei[e]; d = ei[E + e]; } else { s = d = e - E; }
  float v = es[s * H + hd] + ed[d * H + hd];
  v = (v > 0.f) ? v : v * NEG_SLOPE;
  e_buf[t] = v;
  atomicMax(&mbuf[d * H + hd], v);   // global_atomic_max_num_f32
}

// ------------------------------------------- edge pass 2: exp + segment sum
__global__ void edge_exp_sum_kernel(const float* __restrict__ mbuf,
                                    const int* __restrict__ ei,
                                    int E, int n, int H,
                                    float* __restrict__ e_buf,
                                    float* __restrict__ den) {
  int t = blockIdx.x * blockDim.x + threadIdx.x;
  if (t >= (E + n) * H) return;
  const int e = t / H, hd = t % H;
  int d;
  if (e < E) { d = ei[E + e]; } else { d = e - E; }
  float v = __expf(e_buf[t] - mbuf[d * H + hd]);
  e_buf[t] = v;
  atomicAdd(&den[d * H + hd], v);
}

// ---------------------------------- edge pass 3 (layer 1): scatter-aggregate
// One 256-thread block per edge: coalesced h[src] read + agg[dst] atomics.
__global__ __launch_bounds__(256)
void edge_aggregate1_kernel(const float* __restrict__ h1,
                            const float* __restrict__ e_buf,
                            const float* __restrict__ den,
                            const int* __restrict__ ei,
                            int E, int n, float* __restrict__ agg) {
  const int e = blockIdx.x;
  int s, d;
  if (e < E) { s = ei[e]; d = ei[E + e]; } else { s = d = e - E; }
  const int c = threadIdx.x;       // 0..255
  const int hd = c >> 6;           // channel -> head
  const float alpha = e_buf[e * HEADS + hd] / den[d * HEADS + hd];
  atomicAdd(&agg[(size_t)d * F1 + c], h1[(size_t)s * F1 + c] * alpha);
}

// --------------------------------------------------- bias + BatchNorm + ELU
__global__ void bn_elu_kernel(float* __restrict__ xio,
                              const float* __restrict__ b1,
                              const float* __restrict__ gamma,
                              const float* __restrict__ beta,
                              const float* __restrict__ mean,
                              const float* __restrict__ var, int n) {
  long long t = (long long)blockIdx.x * blockDim.x + threadIdx.x;
  if (t >= (long long)n * F1) return;
  const int c = (int)(t % F1);
  float v = xio[t] + b1[c];
  v = (v - mean[c]) * rsqrtf(var[c] + BN_EPS) * gamma[c] + beta[c];
  xio[t] = (v > 0.f) ? v : (__expf(v) - 1.f);
}

// ------------------------------- layer 2 linear (N=2 => scalar, HBM-bound)
__global__ void layer2_lin_kernel(const float* __restrict__ x1,
                                  const float* __restrict__ W2,
                                  const float* __restrict__ a2s,
                                  const float* __restrict__ a2d,
                                  float* __restrict__ h2,
                                  float* __restrict__ es2,
                                  float* __restrict__ ed2, int n) {
  int t = blockIdx.x * blockDim.x + threadIdx.x;
  if (t >= n) return;
  const float* xp = x1 + (size_t)t * F1;
  float o0 = 0.f, o1 = 0.f;
  for (int c = 0; c < F1; ++c) {
    float v = xp[c];
    o0 += v * W2[c * 2 + 0];
    o1 += v * W2[c * 2 + 1];
  }
  h2[t * 2 + 0] = o0;
  h2[t * 2 + 1] = o1;
  es2[t] = o0 * a2s[0] + o1 * a2s[1];
  ed2[t] = o0 * a2d[0] + o1 * a2d[1];
}

// --------------------------- layer 2 aggregate + attention weights to d_out
__global__ void edge_aggregate2_kernel(const float* __restrict__ h2,
                                       const float* __restrict__ ee,
                                       const float* __restrict__ den,
                                       const int* __restrict__ ei,
                                       int E, int n,
                                       float* __restrict__ agg2,
                                       float* __restrict__ att_out) {
  int e = blockIdx.x * blockDim.x + threadIdx.x;
  if (e >= E + n) return;
  int s, d;
  if (e < E) { s = ei[e]; d = ei[E + e]; } else { s = d = e - E; }
  const float alpha = ee[e] / den[d];
  att_out[e] = alpha;                               // att2 [E+n, 1]
  atomicAdd(&agg2[d * 2 + 0], h2[s * 2 + 0] * alpha);
  atomicAdd(&agg2[d * 2 + 1], h2[s * 2 + 1] * alpha);
}

__global__ void finalize_kernel(const float* __restrict__ agg2,
                                const float* __restrict__ b2,
                                float* __restrict__ out, int n) {
  int t = blockIdx.x * blockDim.x + threadIdx.x;
  if (t >= n * 2) return;
  out[t] = agg2[t] + b2[t & 1];
}

// ---------------------------------------------------------------------------
extern "C" void kernel_launch(void* const* d_in, const int* in_sizes, int n_in,
                              void* d_out, int out_size, void* d_ws, size_t ws_size,
                              hipStream_t stream) {
  const float* x     = (const float*)d_in[0];
  const int*   ei    = (const int*)  d_in[1];
  const float* W1    = (const float*)d_in[2];
  const float* a1s   = (const float*)d_in[3];
  const float* a1d   = (const float*)d_in[4];
  const float* b1    = (const float*)d_in[5];
  const float* gamma = (const float*)d_in[6];
  const float* beta  = (const float*)d_in[7];
  const float* mean  = (const float*)d_in[8];
  const float* var   = (const float*)d_in[9];
  const float* W2    = (const float*)d_in[10];
  const float* a2s   = (const float*)d_in[11];
  const float* a2d   = (const float*)d_in[12];
  const float* b2    = (const float*)d_in[13];
  float* out = (float*)d_out;

  const int N  = in_sizes[0] / IN_DIM;   // 50000
  const int E  = in_sizes[1] / 2;        // 400000
  const int ET = E + N;                  // with self loops

  // ---- workspace carve (floats)
  float* w = (float*)d_ws;
  float* h1   = w; w += (size_t)N * F1;
  float* es1  = w; w += (size_t)N * HEADS;
  float* ed1  = w; w += (size_t)N * HEADS;
  float* m1   = w; w += (size_t)N * HEADS;
  float* den1 = w; w += (size_t)N * HEADS;
  float* e1   = w; w += (size_t)ET * HEADS;
  float* agg1 = w; w += (size_t)N * F1;   // becomes x1 in-place after BN/ELU
  float* h2   = w; w += (size_t)N * 2;
  float* es2  = w; w += (size_t)N;
  float* ed2  = w; w += (size_t)N;
  float* m2   = w; w += (size_t)N;
  float* den2 = w; w += (size_t)N;
  float* ee2  = w; w += (size_t)ET;
  float* agg2 = w; w += (size_t)N * 2;

  auto blocks = [](long long n, int b) { return (unsigned)((n + b - 1) / b); };

  // ---- re-init accumulators every call (graph-replay safe)
  fill_kernel<<<blocks((long long)N * HEADS, 256), 256, 0, stream>>>(m1,   (long long)N * HEADS, -INFINITY);
  fill_kernel<<<blocks((long long)N * HEADS, 256), 256, 0, stream>>>(den1, (long long)N * HEADS, 0.f);
  fill_kernel<<<blocks((long long)N * F1,   256), 256, 0, stream>>>(agg1, (long long)N * F1,    0.f);
  fill_kernel<<<blocks((long long)N,        256), 256, 0, stream>>>(m2,   (long long)N,         -INFINITY);
  fill_kernel<<<blocks((long long)N,        256), 256, 0, stream>>>(den2, (long long)N,         0.f);
  fill_kernel<<<blocks((long long)N * 2,    256), 256, 0, stream>>>(agg2, (long long)N * 2,     0.f);

  // ---- layer 1
  gemm1_wmma_kernel<<<blocks(N, ROWS_PB), 256, 0, stream>>>(x, W1, h1, N);
  att_logits1_kernel<<<blocks((long long)N * HEADS, 256), 256, 0, stream>>>(h1, a1s, a1d, es1, ed1, N);
  edge_logits_max_kernel<<<blocks((long long)ET * HEADS, 256), 256, 0, stream>>>(es1, ed1, ei, E, N, HEADS, e1, m1);
  edge_exp_sum_kernel<<<blocks((long long)ET * HEADS, 256), 256, 0, stream>>>(m1, ei, E, N, HEADS, e1, den1);
  edge_aggregate1_kernel<<<(unsigned)ET, 256, 0, stream>>>(h1, e1, den1, ei, E, N, agg1);
  bn_elu_kernel<<<blocks((long long)N * F1, 256), 256, 0, stream>>>(agg1, b1, gamma, beta, mean, var, N);

  // ---- layer 2
  layer2_lin_kernel<<<blocks(N, 256), 256, 0, stream>>>(agg1, W2, a2s, a2d, h2, es2, ed2, N);
  edge_logits_max_kernel<<<blocks(ET, 256), 256, 0, stream>>>(es2, ed2, ei, E, N, 1, ee2, m2);
  edge_exp_sum_kernel<<<blocks(ET, 256), 256, 0, stream>>>(m2, ei, E, N, 1, ee2, den2);
  edge_aggregate2_kernel<<<blocks(ET, 256), 256, 0, stream>>>(h2, ee2, den2, ei, E, N, agg2,
                                                              out + (size_t)2 * N);
  finalize_kernel<<<blocks((long long)N * 2, 256), 256, 0, stream>>>(agg2, b2, out, N);
}